// EdgeDetectionPromote_32873679683961
// MI455X (gfx1250) — compile-verified
//
#include <hip/hip_runtime.h>
#include <cstdint>

// Problem geometry (fixed by the reference): x = [32, 3, 512, 512] f32 NCHW,
// out = [32, 1, 512, 512] f32 edge map.
constexpr int IMG_H = 512;
constexpr int IMG_W = 512;
constexpr int CH    = 3;

// Tiling: 64x16 output pixels per block, 256 threads (8 wave32).
constexpr int TILE_W  = 64;
constexpr int TILE_H  = 16;
constexpr int LDS_W   = TILE_W + 2;   // 66 (1-wide halo each side, zero pad)
constexpr int LDS_H   = TILE_H + 2;   // 18
constexpr int LDS_STR = 67;           // 67 mod 64 = 3 -> staggers row bank classes
constexpr int NTHREADS = 256;

__global__ __launch_bounds__(NTHREADS)
void sobel_edge_kernel(const float* __restrict__ x, float* __restrict__ out)
{
    __shared__ float tile[CH * LDS_H * LDS_STR];

    const int tid = threadIdx.x;
    const int tx0 = blockIdx.x * TILE_W;
    const int ty0 = blockIdx.y * TILE_H;
    const int b   = blockIdx.z;

    // Wave-relative LDS byte offset of the tile (generic ptr low 32 bits are
    // the LDS offset per the gfx1250 flat-aperture mapping).
    const uint32_t lds_base = (uint32_t)(uintptr_t)(&tile[0]);
    const float* xb = x + (size_t)b * CH * IMG_H * IMG_W;

    // ---- Stage halo tile into LDS via CDNA5 async global->LDS DMA path ----
    // 3 * 18 * 66 = 3564 elements, ~14 per thread. In-bounds elements use
    // global_load_async_to_lds_b32 (ASYNCcnt); out-of-bounds halo slots are
    // zeroed with a plain LDS store (disjoint addresses -> no race).
    for (int i = tid; i < CH * LDS_H * LDS_W; i += NTHREADS) {
        int c   = i / (LDS_H * LDS_W);
        int rem = i - c * (LDS_H * LDS_W);
        int ly  = rem / LDS_W;
        int lx  = rem - ly * LDS_W;
        int gy  = ty0 - 1 + ly;
        int gx  = tx0 - 1 + lx;
        uint32_t lds_addr = lds_base +
            4u * (uint32_t)(c * (LDS_H * LDS_STR) + ly * LDS_STR + lx);
        if ((unsigned)gy < (unsigned)IMG_H && (unsigned)gx < (unsigned)IMG_W) {
            const float* gp = xb + (size_t)c * (IMG_H * IMG_W)
                                 + (size_t)gy * IMG_W + gx;
            asm volatile("global_load_async_to_lds_b32 %0, %1, off"
                         :: "v"(lds_addr), "v"(gp)
                         : "memory");
        } else {
            tile[c * (LDS_H * LDS_STR) + ly * LDS_STR + lx] = 0.0f;
        }
    }

    // Drain this wave's async copies, then sync the workgroup.
    asm volatile("s_wait_asynccnt 0" ::: "memory");
    __syncthreads();

    // ---- Compute: 4 consecutive output pixels per thread ----
    const int tx4 = (tid & 15) * 4;   // column of first pixel within tile
    const int ty  = tid >> 4;         // row within tile (0..15)

    bool e0 = false, e1 = false, e2 = false, e3 = false;

    #pragma unroll
    for (int c = 0; c < CH; ++c) {
        const float* s  = &tile[c * (LDS_H * LDS_STR)];
        const float* rt = s + (ty + 0) * LDS_STR + tx4;  // top row
        const float* rm = s + (ty + 1) * LDS_STR + tx4;  // mid row
        const float* rb = s + (ty + 2) * LDS_STR + tx4;  // bottom row

        float t[6], m[6], bo[6];
        #pragma unroll
        for (int k = 0; k < 6; ++k) { t[k] = rt[k]; m[k] = rm[k]; bo[k] = rb[k]; }

        #pragma unroll
        for (int p = 0; p < 4; ++p) {
            // Sobel (cross-correlation, matching jax.lax.conv)
            float gxv = (t[p + 2] - t[p]) + 2.0f * (m[p + 2] - m[p])
                      + (bo[p + 2] - bo[p]);
            float gyv = (bo[p] + 2.0f * bo[p + 1] + bo[p + 2])
                      - (t[p]  + 2.0f * t[p + 1]  + t[p + 2]);
            // mag > 0.2  <=>  mag^2 > 0.04 (both sides non-negative)
            bool e = (gxv * gxv + gyv * gyv) > 0.04f;
            if      (p == 0) e0 |= e;
            else if (p == 1) e1 |= e;
            else if (p == 2) e2 |= e;
            else             e3 |= e;
        }
    }

    float4 r;
    r.x = e0 ? 1.0f : 0.0f;
    r.y = e1 ? 1.0f : 0.0f;
    r.z = e2 ? 1.0f : 0.0f;
    r.w = e3 ? 1.0f : 0.0f;

    float* op = out + (size_t)b * (IMG_H * IMG_W)
                    + (size_t)(ty0 + ty) * IMG_W + (tx0 + tx4);
    *reinterpret_cast<float4*>(op) = r;   // 16B-aligned coalesced store
}

extern "C" void kernel_launch(void* const* d_in, const int* in_sizes, int n_in,
                              void* d_out, int out_size, void* d_ws, size_t ws_size,
                              hipStream_t stream) {
    const float* x = (const float*)d_in[0];
    float* out = (float*)d_out;

    const int batch = in_sizes[0] / (CH * IMG_H * IMG_W);  // 32
    dim3 grid(IMG_W / TILE_W, IMG_H / TILE_H, batch);      // 8 x 32 x 32
    dim3 block(NTHREADS);
    hipLaunchKernelGGL(sobel_edge_kernel, grid, block, 0, stream, x, out);
}